// MoEGate_18047452578718
// MI455X (gfx1250) — compile-verified
//
#include <hip/hip_runtime.h>

// ---------------------------------------------------------------------------
// MoE gate for MI455X (gfx1250, wave32, WMMA).
// logits = x @ W^T via V_WMMA_F32_16X16X4_F32, then softmax/top2/aux-loss.
// Software-pipelined K loop: loads for k+8 issued before WMMAs consuming k.
// ---------------------------------------------------------------------------

typedef __attribute__((ext_vector_type(2))) float v2f;
typedef __attribute__((ext_vector_type(4))) float v4f;
typedef __attribute__((ext_vector_type(8))) float v8f;

#define HDIM 2048
#define TTOK 16384           // 4 * 4096 tokens
#define NEXP 8
#define TILES_PER_BLOCK 8    // 8 waves/block, one 16-token tile per wave
#define NBLOCKS (TTOK / 16 / TILES_PER_BLOCK)   // 128

__device__ __forceinline__ v8f wmma_step(v8f acc, v4f a, v4f b) {
    v2f a0 = {a.x, a.y}, a1 = {a.z, a.w};
    v2f b0 = {b.x, b.y}, b1 = {b.z, b.w};
    acc = __builtin_amdgcn_wmma_f32_16x16x4_f32(false, a0, false, b0,
                                                (short)0, acc, false, false);
    acc = __builtin_amdgcn_wmma_f32_16x16x4_f32(false, a1, false, b1,
                                                (short)0, acc, false, false);
    return acc;
}

__global__ __launch_bounds__(256) void moe_gate_main(
    const float* __restrict__ x,      // [TTOK, HDIM]
    const float* __restrict__ w,      // [NEXP, HDIM]
    float* __restrict__ out,          // [2T idx | 2T weights | 1 loss]
    float* __restrict__ partials)     // [NBLOCKS, 16] in d_ws
{
    __shared__ float lg[TILES_PER_BLOCK][16 * 9];   // per-wave logits, padded
    __shared__ float wsum[TILES_PER_BLOCK][8];
    __shared__ float wcnt[TILES_PER_BLOCK][8];

    const int lane = threadIdx.x & 31;
    const int wave = threadIdx.x >> 5;
    const int tile = blockIdx.x * TILES_PER_BLOCK + wave;
    const int hi   = lane >> 4;        // 0: lo half (K k..k+3), 1: hi half (K k+4..k+7)
    const int m    = lane & 15;        // A-row (token within tile) / B-col (expert)
    const int ew   = m & 7;            // safe weight row for pad cols 8..15

    const float* xp = x + (size_t)(tile * 16 + m) * HDIM + 4 * hi;
    const float* wp = w + (size_t)ew * HDIM + 4 * hi;

    // ---- GEMM: acc = 16x16 f32 tile of logits (cols 0..7 valid) ----------
    // One-stage software pipeline: the b128 loads for k+8 are in flight while
    // the two WMMAs for k execute (compiler can wait loadcnt<=2, not 0).
    v8f acc = {};
    v4f a_cur = __builtin_nontemporal_load((const v4f*)xp);
    v4f b_cur = *(const v4f*)wp;

    for (int k = 0; k < HDIM - 32; k += 32) {
        // speculative L2 prefetch ~4 KB ahead, once per 128 B consumed
        __builtin_prefetch(xp + k + 1056, 0, 0);
        #pragma unroll
        for (int u = 0; u < 4; ++u) {
            v4f a_nxt = __builtin_nontemporal_load((const v4f*)(xp + k + 8 * u + 8));
            v4f b_nxt = *(const v4f*)(wp + k + 8 * u + 8);
            acc = wmma_step(acc, a_cur, b_cur);
            a_cur = a_nxt; b_cur = b_nxt;
        }
    }
    // epilogue: last 32 floats (a_cur currently holds k = HDIM-32)
    #pragma unroll
    for (int u = 0; u < 3; ++u) {
        v4f a_nxt = __builtin_nontemporal_load((const v4f*)(xp + (HDIM - 24) + 8 * u));
        v4f b_nxt = *(const v4f*)(wp + (HDIM - 24) + 8 * u);
        acc = wmma_step(acc, a_cur, b_cur);
        a_cur = a_nxt; b_cur = b_nxt;
    }
    acc = wmma_step(acc, a_cur, b_cur);

    // ---- transpose C layout to [token][expert] via LDS --------------------
    // lane holds expert col m, VGPR r holds token row r + 8*hi.
    if (m < NEXP) {
        #pragma unroll
        for (int r = 0; r < 8; ++r)
            lg[wave][(r + 8 * hi) * 9 + m] = acc[r];
    }
    __syncthreads();

    // ---- per-token softmax + top-2 (lanes 0..15 each own one token) -------
    float sc[8], ct[8];
    if (lane < 16) {
        const float* L = &lg[wave][lane * 9];
        float v[8], mx = L[0];
        #pragma unroll
        for (int e = 0; e < 8; ++e) { v[e] = L[e]; mx = fmaxf(mx, v[e]); }
        float sum = 0.f;
        #pragma unroll
        for (int e = 0; e < 8; ++e) { sc[e] = __expf(v[e] - mx); sum += sc[e]; }
        float inv = 1.0f / sum;
        #pragma unroll
        for (int e = 0; e < 8; ++e) sc[e] *= inv;

        int i1 = 0; float p1 = sc[0];
        #pragma unroll
        for (int e = 1; e < 8; ++e) if (sc[e] > p1) { p1 = sc[e]; i1 = e; }
        int i2 = -1; float p2 = -1.f;
        #pragma unroll
        for (int e = 0; e < 8; ++e)
            if (e != i1 && sc[e] > p2) { p2 = sc[e]; i2 = e; }

        const int tok = tile * 16 + lane;
        out[2 * tok]                 = (float)i1;
        out[2 * tok + 1]             = (float)i2;
        out[2 * TTOK + 2 * tok]      = p1;
        out[2 * TTOK + 2 * tok + 1]  = p2;
        #pragma unroll
        for (int e = 0; e < 8; ++e)
            ct[e] = (e == i1 ? 1.f : 0.f) + (e == i2 ? 1.f : 0.f);
    } else {
        #pragma unroll
        for (int e = 0; e < 8; ++e) { sc[e] = 0.f; ct[e] = 0.f; }
    }

    // ---- deterministic wave reduction -> LDS -> per-block partials --------
    #pragma unroll
    for (int e = 0; e < 8; ++e) {
        float v = sc[e], cv = ct[e];
        for (int off = 16; off > 0; off >>= 1) {
            v  += __shfl_down(v,  off, 32);
            cv += __shfl_down(cv, off, 32);
        }
        if (lane == 0) { wsum[wave][e] = v; wcnt[wave][e] = cv; }
    }
    __syncthreads();

    if (threadIdx.x < 8) {
        float s = 0.f;
        for (int q = 0; q < TILES_PER_BLOCK; ++q) s += wsum[q][threadIdx.x];
        partials[blockIdx.x * 16 + threadIdx.x] = s;
    } else if (threadIdx.x < 16) {
        const int e = threadIdx.x - 8;
        float s = 0.f;
        for (int q = 0; q < TILES_PER_BLOCK; ++q) s += wcnt[q][e];
        partials[blockIdx.x * 16 + 8 + e] = s;
    }
}

__global__ void moe_gate_finalize(const float* __restrict__ partials,
                                  float* __restrict__ out)
{
    if (threadIdx.x == 0 && blockIdx.x == 0) {
        float S[8] = {}, C[8] = {};
        for (int b = 0; b < NBLOCKS; ++b)
            for (int e = 0; e < 8; ++e) {
                S[e] += partials[b * 16 + e];
                C[e] += partials[b * 16 + 8 + e];
            }
        float aux = 0.f;
        for (int e = 0; e < 8; ++e) {
            float Pi = S[e] / (float)TTOK;                 // mean score
            float ce = C[e] / (float)(TTOK * 2);           // count fraction
            aux += Pi * (ce * (float)NEXP);                // Pi * fi
        }
        out[4 * TTOK] = aux * 0.01f;                       // ALPHA
    }
}

extern "C" void kernel_launch(void* const* d_in, const int* in_sizes, int n_in,
                              void* d_out, int out_size, void* d_ws, size_t ws_size,
                              hipStream_t stream) {
    const float* x = (const float*)d_in[0];   // [4,4096,2048] f32
    const float* w = (const float*)d_in[1];   // [8,2048]     f32
    float* out      = (float*)d_out;          // 65537 f32
    float* partials = (float*)d_ws;           // 128*16 f32 = 8 KB scratch

    moe_gate_main<<<NBLOCKS, 256, 0, stream>>>(x, w, out, partials);
    moe_gate_finalize<<<1, 32, 0, stream>>>(partials, out);
}